// SimPGCN_12463995093672
// MI455X (gfx1250) — compile-verified
//
#include <hip/hip_runtime.h>
#include <stdint.h>

typedef __attribute__((ext_vector_type(2))) float v2f;
typedef __attribute__((ext_vector_type(8))) float v8f;

#define GAMMA 0.1f

// ---------------------------------------------------------------- utilities
__global__ void zero_u32(unsigned* __restrict__ p, long long n) {
    long long i = (long long)blockIdx.x * blockDim.x + threadIdx.x;
    if (i < n) p[i] = 0u;
}

__global__ void copy_i32(const int* __restrict__ a, int* __restrict__ b, int n) {
    int i = blockIdx.x * blockDim.x + threadIdx.x;
    if (i < n) b[i] = a[i];
}

// integer degree counts (also reused as CSR segment lengths)
__global__ void count_kernel(const int* __restrict__ src, const int* __restrict__ dst,
                             int* __restrict__ cout, int* __restrict__ cin, int E) {
    int e = blockIdx.x * blockDim.x + threadIdx.x;
    if (e >= E) return;
    atomicAdd(&cout[src[e]], 1);
    atomicAdd(&cin[dst[e]], 1);
}

// norm = (deg+add > 0) ? (deg+add)^-1/2 : 1      (add=1 models the self loop)
__global__ void norm_from_cnt(const int* __restrict__ cnt, float* __restrict__ nrm,
                              float add, int N) {
    int i = blockIdx.x * blockDim.x + threadIdx.x;
    if (i >= N) return;
    float dg = (float)cnt[i] + add;
    nrm[i] = (dg > 0.0f) ? rsqrtf(dg) : 1.0f;
}

// ------------------------------------------------------- exclusive scan (3-phase)
__global__ void scan256(const int* __restrict__ in, int* __restrict__ excl,
                        int* __restrict__ bsum, int N) {
    __shared__ int tmp[256];
    int i = blockIdx.x * 256 + threadIdx.x;
    int v = (i < N) ? in[i] : 0;
    tmp[threadIdx.x] = v;
    __syncthreads();
    for (int off = 1; off < 256; off <<= 1) {
        int a = (threadIdx.x >= off) ? tmp[threadIdx.x - off] : 0;
        __syncthreads();
        tmp[threadIdx.x] += a;
        __syncthreads();
    }
    if (i < N) excl[i] = tmp[threadIdx.x] - v;
    if (threadIdx.x == 255) bsum[blockIdx.x] = tmp[255];
}

__global__ void scan_bsum(int* __restrict__ bsum, int nb) {  // nb <= 1024
    __shared__ int tmp[1024];
    int t = threadIdx.x;
    int v = (t < nb) ? bsum[t] : 0;
    tmp[t] = v;
    __syncthreads();
    for (int off = 1; off < 1024; off <<= 1) {
        int a = (t >= off) ? tmp[t - off] : 0;
        __syncthreads();
        tmp[t] += a;
        __syncthreads();
    }
    if (t < nb) bsum[t] = tmp[t] - v;  // exclusive
}

__global__ void scan_finish(int* __restrict__ excl, const int* __restrict__ bsum, int N) {
    int i = blockIdx.x * 256 + threadIdx.x;
    if (i < N) excl[i] += bsum[i >> 8];
}

// scatter edges into dst-sorted CSR, with precomputed source weight no[src]
__global__ void csr_fill(const int* __restrict__ src, const int* __restrict__ dst,
                         const float* __restrict__ no, int* __restrict__ cursor,
                         int* __restrict__ csr_src, float* __restrict__ csr_w, int E) {
    int e = blockIdx.x * blockDim.x + threadIdx.x;
    if (e >= E) return;
    int s = src[e], d = dst[e];
    int p = atomicAdd(&cursor[d], 1);
    csr_src[p] = s;
    csr_w[p] = no[s];
}

// ------------------------------------------------------------ per-row gates
// one wave32 per row: s = sigmoid(x.sc + b), d = x.dk + db
__global__ void gate_kernel(const float* __restrict__ X, const float* __restrict__ sc,
                            const float* __restrict__ b, const float* __restrict__ dk,
                            const float* __restrict__ db, float* __restrict__ sOut,
                            float* __restrict__ dOut, int N, int F) {
    int gtid = blockIdx.x * blockDim.x + threadIdx.x;
    int row  = gtid >> 5;
    int lane = threadIdx.x & 31;
    if (row >= N) return;
    const float* xr = X + (long long)row * F;
    float as = 0.0f, ad = 0.0f;
    for (int j = lane; j < F; j += 32) {
        float v = xr[j];
        as += v * sc[j];
        ad += v * dk[j];
    }
    #pragma unroll
    for (int off = 16; off; off >>= 1) {
        as += __shfl_xor(as, off, 32);
        ad += __shfl_xor(ad, off, 32);
    }
    if (lane == 0) {
        sOut[row] = 1.0f / (1.0f + expf(-(as + b[0])));
        dOut[row] = ad + db[0];
    }
}

// --------------------------------------------------- GEMM1: [N,512]x[512,64]
// Double-buffered LDS tiles filled with global_load_async_to_lds_b128
// (ASYNCcnt), consumed by f32 WMMA 16x16x4. Block = 8 waves = 32x64 tile.
#define SA_STRIDE 68   // 68*4 bytes: 16B aligned rows, conflict-free A reads
#define SB_STRIDE 80   // 80*4 bytes: 16B aligned rows, conflict-free B reads

__device__ __forceinline__ void stage_async(const float* __restrict__ A,
                                            const float* __restrict__ B,
                                            unsigned ldsA, unsigned ldsB,
                                            int row0, int kc, int t, int N) {
    // A tile: 32 rows x 64 k = 512 chunks of 16B, 2 per thread
    #pragma unroll
    for (int j = 0; j < 2; ++j) {
        int idx = t + j * 256;
        int r = idx >> 4, c4 = (idx & 15) * 4;
        int gr = row0 + r;
        if (gr > N - 1) gr = N - 1;  // clamp: padded rows discarded at store
        const float* g = A + (long long)gr * 512 + kc + c4;
        unsigned l = ldsA + (unsigned)((r * SA_STRIDE + c4) * 4);
        asm volatile("global_load_async_to_lds_b128 %0, %1, off"
                     :: "v"(l), "v"(g) : "memory");
    }
    // B tile: 64 k x 64 cols = 1024 chunks of 16B, 4 per thread
    #pragma unroll
    for (int j = 0; j < 4; ++j) {
        int idx = t + j * 256;
        int r = idx >> 4, c4 = (idx & 15) * 4;
        const float* g = B + (long long)(kc + r) * 64 + c4;
        unsigned l = ldsB + (unsigned)((r * SB_STRIDE + c4) * 4);
        asm volatile("global_load_async_to_lds_b128 %0, %1, off"
                     :: "v"(l), "v"(g) : "memory");
    }
}

__global__ __launch_bounds__(256) void gemm1_wmma(const float* __restrict__ A,
                                                  const float* __restrict__ B,
                                                  float* __restrict__ X, int N) {
    __shared__ float sA[2][32 * SA_STRIDE];
    __shared__ float sB[2][64 * SB_STRIDE];
    const int t     = threadIdx.x;
    const int wave  = t >> 5;
    const int lane  = t & 31;
    const int rt    = wave >> 2;        // 0..1 : row tile
    const int ct    = wave & 3;         // 0..3 : col tile
    const int row0  = blockIdx.x * 32;
    const int l15   = lane & 15;
    const int khalf = (lane & 16) ? 2 : 0;
    const int mbase = (lane & 16) ? 8 : 0;

    const unsigned uA[2] = { (unsigned)(uintptr_t)&sA[0][0], (unsigned)(uintptr_t)&sA[1][0] };
    const unsigned uB[2] = { (unsigned)(uintptr_t)&sB[0][0], (unsigned)(uintptr_t)&sB[1][0] };

    stage_async(A, B, uA[0], uB[0], row0, 0, t, N);   // prologue: buffer 0

    v8f acc = {};
    for (int kc = 0; kc < 512; kc += 64) {
        const int cur = (kc >> 6) & 1;
        asm volatile("s_wait_asynccnt 0x0" ::: "memory");
        __syncthreads();                               // buf[cur] fully landed
        if (kc + 64 < 512)                             // prefetch next chunk
            stage_async(A, B, uA[cur ^ 1], uB[cur ^ 1], row0, kc + 64, t, N);
        const float* sAc = &sA[cur][0];
        const float* sBc = &sB[cur][0];
        #pragma unroll
        for (int kk = 0; kk < 64; kk += 4) {
            v2f a, b;
            a.x = sAc[(rt * 16 + l15) * SA_STRIDE + kk + khalf];
            a.y = sAc[(rt * 16 + l15) * SA_STRIDE + kk + khalf + 1];
            b.x = sBc[(kk + khalf) * SB_STRIDE + ct * 16 + l15];
            b.y = sBc[(kk + khalf + 1) * SB_STRIDE + ct * 16 + l15];
            acc = __builtin_amdgcn_wmma_f32_16x16x4_f32(false, a, false, b,
                                                        (short)0, acc, false, false);
        }
        __syncthreads();   // reads of buf[cur] done before it is refilled
    }
    #pragma unroll
    for (int v = 0; v < 8; ++v) {
        int gr = row0 + rt * 16 + mbase + v;
        if (gr < N) X[(long long)gr * 64 + ct * 16 + l15] = acc[v];
    }
}

// ------------------------------------------------ GEMM2: [N,64]x[64,10(->16)]
__global__ __launch_bounds__(256) void gemm2_wmma(const float* __restrict__ H,
                                                  const float* __restrict__ W2,
                                                  float* __restrict__ X2, int N) {
    __shared__ float sB[64 * 16];
    const int t = threadIdx.x;
    #pragma unroll
    for (int i = 0; i < 4; ++i) {
        int idx = t + i * 256;
        int r = idx >> 4, c = idx & 15;
        sB[idx] = (c < 10) ? W2[r * 10 + c] : 0.0f;
    }
    __syncthreads();
    const int wave  = t >> 5;
    const int lane  = t & 31;
    const int l15   = lane & 15;
    const int khalf = (lane & 16) ? 2 : 0;
    const int mbase = (lane & 16) ? 8 : 0;
    const int row0  = blockIdx.x * 128 + wave * 16;
    const int arow  = row0 + l15;

    v8f acc = {};
    #pragma unroll
    for (int kk = 0; kk < 64; kk += 4) {
        v2f a, b;
        a.x = (arow < N) ? H[(long long)arow * 64 + kk + khalf] : 0.0f;
        a.y = (arow < N) ? H[(long long)arow * 64 + kk + khalf + 1] : 0.0f;
        b.x = sB[(kk + khalf) * 16 + l15];
        b.y = sB[(kk + khalf + 1) * 16 + l15];
        acc = __builtin_amdgcn_wmma_f32_16x16x4_f32(false, a, false, b,
                                                    (short)0, acc, false, false);
    }
    if (l15 < 10) {
        #pragma unroll
        for (int v = 0; v < 8; ++v) {
            int gr = row0 + mbase + v;
            if (gr < N) X2[(long long)gr * 10 + l15] = acc[v];
        }
    }
}

// ------------------------------------------------- CSR gather aggregation
// D=64: one wave32 per node, 2 feats/lane, register accumulation, no atomics
__global__ void gather64(const float* __restrict__ x, const int* __restrict__ csr_src,
                         const float* __restrict__ csr_w, const int* __restrict__ offs,
                         const int* __restrict__ cnt, float* __restrict__ agg, int N) {
    int gtid = blockIdx.x * blockDim.x + threadIdx.x;
    int node = gtid >> 5;
    int lane = threadIdx.x & 31;
    if (node >= N) return;
    int beg = offs[node];
    int end = beg + cnt[node];
    float a0 = 0.0f, a1 = 0.0f;
    for (int p = beg; p < end; ++p) {
        int s = csr_src[p];
        float w = csr_w[p];
        const float* xs = x + (long long)s * 64;
        a0 += xs[lane] * w;
        a1 += xs[lane + 32] * w;
    }
    float* ag = agg + (long long)node * 64;
    ag[lane] = a0;
    ag[lane + 32] = a1;
}

// D=10: 16 lanes per node (2 nodes/wave)
__global__ void gather10(const float* __restrict__ x, const int* __restrict__ csr_src,
                         const float* __restrict__ csr_w, const int* __restrict__ offs,
                         const int* __restrict__ cnt, float* __restrict__ agg, int N) {
    int gtid = blockIdx.x * blockDim.x + threadIdx.x;
    int node = gtid >> 4;
    int f = gtid & 15;
    if (node >= N || f >= 10) return;
    int beg = offs[node];
    int end = beg + cnt[node];
    float a = 0.0f;
    for (int p = beg; p < end; ++p)
        a += x[(long long)csr_src[p] * 10 + f] * csr_w[p];
    agg[(long long)node * 10 + f] = a;
}

// --------------------------------------------------------------- combines
// out = s*ni_o*(aggO + x*no_o) + (1-s)*ni_k*aggK + GAMMA*d*x  (self loop folded)
__global__ void combine64(const float* __restrict__ x, const float* __restrict__ aggO,
                          const float* __restrict__ aggK, const float* __restrict__ sArr,
                          const float* __restrict__ dArr, const float* __restrict__ ni_o,
                          const float* __restrict__ no_o, const float* __restrict__ ni_k,
                          float* __restrict__ out, int N) {
    long long idx = (long long)blockIdx.x * blockDim.x + threadIdx.x;
    if (idx >= (long long)N * 64) return;
    int i = (int)(idx >> 6);
    float xv = x[idx];
    float ao = (aggO[idx] + xv * no_o[i]) * ni_o[i];
    float ak = aggK[idx] * ni_k[i];
    float si = sArr[i];
    out[idx] = si * ao + (1.0f - si) * ak + GAMMA * dArr[i] * xv;
}

__global__ void combine10(const float* __restrict__ x, const float* __restrict__ aggO,
                          const float* __restrict__ aggK, const float* __restrict__ sArr,
                          const float* __restrict__ dArr, const float* __restrict__ ni_o,
                          const float* __restrict__ no_o, const float* __restrict__ ni_k,
                          float* __restrict__ out, int N) {
    long long idx = (long long)blockIdx.x * blockDim.x + threadIdx.x;
    if (idx >= (long long)N * 10) return;
    int i = (int)(idx / 10);
    float xv = x[idx];
    float ao = (aggO[idx] + xv * no_o[i]) * ni_o[i];
    float ak = aggK[idx] * ni_k[i];
    float si = sArr[i];
    out[idx] = si * ao + (1.0f - si) * ak + GAMMA * dArr[i] * xv;
}

// ================================================================ launcher
extern "C" void kernel_launch(void* const* d_in, const int* in_sizes, int n_in,
                              void* d_out, int out_size, void* d_ws, size_t ws_size,
                              hipStream_t stream) {
    const float* feat    = (const float*)d_in[0];
    const float* W1      = (const float*)d_in[1];
    const float* W2      = (const float*)d_in[2];
    const float* scores1 = (const float*)d_in[3];
    const float* scores2 = (const float*)d_in[4];
    const float* b1      = (const float*)d_in[5];
    const float* b2      = (const float*)d_in[6];
    const float* Dk1     = (const float*)d_in[7];
    const float* Dk2     = (const float*)d_in[8];
    const float* db1     = (const float*)d_in[9];
    const float* db2     = (const float*)d_in[10];
    const int* edge_src  = (const int*)d_in[11];
    const int* edge_dst  = (const int*)d_in[12];
    const int* knn_src   = (const int*)d_in[13];
    const int* knn_dst   = (const int*)d_in[14];
    float* out = (float*)d_out;

    const int F_IN = in_sizes[3];            // 512
    const int N    = in_sizes[0] / F_IN;     // 100000
    const int E    = in_sizes[11];           // 1.6M
    const int EK   = in_sizes[13];           // 1.0M

    // -------- workspace layout (4-byte slots) --------
    size_t off4 = 0;
    auto alloc = [&](size_t n4) { void* p = (char*)d_ws + off4 * 4; off4 += n4; return p; };
    float* ni_o  = (float*)alloc(N);
    float* no_o  = (float*)alloc(N);
    float* ni_k  = (float*)alloc(N);
    float* no_k  = (float*)alloc(N);
    float* sbuf  = (float*)alloc(N);
    float* dbuf  = (float*)alloc(N);
    int* cin_o   = (int*)alloc(N);
    int* cout_o  = (int*)alloc(N);
    int* cin_k   = (int*)alloc(N);
    int* cout_k  = (int*)alloc(N);
    int* offs_o  = (int*)alloc(N);
    int* offs_k  = (int*)alloc(N);
    int* cur_o   = (int*)alloc(N);
    int* cur_k   = (int*)alloc(N);
    int* bsum    = (int*)alloc(1024);
    int* csrs_o  = (int*)alloc(E);
    float* csrw_o = (float*)alloc(E);
    int* csrs_k  = (int*)alloc(EK);
    float* csrw_k = (float*)alloc(EK);
    float* x1    = (float*)alloc((size_t)N * 64);
    float* aggO  = (float*)alloc((size_t)N * 64);
    float* aggK  = (float*)alloc((size_t)N * 64);
    float* hbuf  = (float*)alloc((size_t)N * 64);
    float* x2    = x1;    // layer-2 reuse
    float* aggO2 = aggO;
    float* aggK2 = aggK;

    const int T = 256;
    const int nb = (N + 255) / 256;
    auto blk = [](long long n, int t) { return (int)((n + t - 1) / t); };

    // -------- degrees, norms, CSR (shared by both layers) --------
    zero_u32<<<blk((long long)4 * N, T), T, 0, stream>>>((unsigned*)cin_o, (long long)4 * N);
    count_kernel<<<blk(E, T), T, 0, stream>>>(edge_src, edge_dst, cout_o, cin_o, E);
    count_kernel<<<blk(EK, T), T, 0, stream>>>(knn_src, knn_dst, cout_k, cin_k, EK);
    norm_from_cnt<<<blk(N, T), T, 0, stream>>>(cin_o, ni_o, 1.0f, N);
    norm_from_cnt<<<blk(N, T), T, 0, stream>>>(cout_o, no_o, 1.0f, N);
    norm_from_cnt<<<blk(N, T), T, 0, stream>>>(cin_k, ni_k, 0.0f, N);
    norm_from_cnt<<<blk(N, T), T, 0, stream>>>(cout_k, no_k, 0.0f, N);

    // CSR for original graph (dst-sorted)
    scan256<<<nb, 256, 0, stream>>>(cin_o, offs_o, bsum, N);
    scan_bsum<<<1, 1024, 0, stream>>>(bsum, nb);
    scan_finish<<<nb, 256, 0, stream>>>(offs_o, bsum, N);
    copy_i32<<<blk(N, T), T, 0, stream>>>(offs_o, cur_o, N);
    csr_fill<<<blk(E, T), T, 0, stream>>>(edge_src, edge_dst, no_o, cur_o, csrs_o, csrw_o, E);
    // CSR for knn graph
    scan256<<<nb, 256, 0, stream>>>(cin_k, offs_k, bsum, N);
    scan_bsum<<<1, 1024, 0, stream>>>(bsum, nb);
    scan_finish<<<nb, 256, 0, stream>>>(offs_k, bsum, N);
    copy_i32<<<blk(N, T), T, 0, stream>>>(offs_k, cur_k, N);
    csr_fill<<<blk(EK, T), T, 0, stream>>>(knn_src, knn_dst, no_k, cur_k, csrs_k, csrw_k, EK);

    // ================= layer 1 =================
    gate_kernel<<<blk((long long)N * 32, T), T, 0, stream>>>(
        feat, scores1, b1, Dk1, db1, sbuf, dbuf, N, 512);
    gemm1_wmma<<<blk(N, 32), T, 0, stream>>>(feat, W1, x1, N);

    gather64<<<blk((long long)N * 32, T), T, 0, stream>>>(x1, csrs_o, csrw_o, offs_o, cin_o, aggO, N);
    gather64<<<blk((long long)N * 32, T), T, 0, stream>>>(x1, csrs_k, csrw_k, offs_k, cin_k, aggK, N);
    combine64<<<blk((long long)N * 64, T), T, 0, stream>>>(
        x1, aggO, aggK, sbuf, dbuf, ni_o, no_o, ni_k, hbuf, N);

    // ================= layer 2 =================
    gate_kernel<<<blk((long long)N * 32, T), T, 0, stream>>>(
        hbuf, scores2, b2, Dk2, db2, sbuf, dbuf, N, 64);
    gemm2_wmma<<<blk(N, 128), T, 0, stream>>>(hbuf, W2, x2, N);

    gather10<<<blk((long long)N * 16, T), T, 0, stream>>>(x2, csrs_o, csrw_o, offs_o, cin_o, aggO2, N);
    gather10<<<blk((long long)N * 16, T), T, 0, stream>>>(x2, csrs_k, csrw_k, offs_k, cin_k, aggK2, N);
    combine10<<<blk((long long)N * 10, T), T, 0, stream>>>(
        x2, aggO2, aggK2, sbuf, dbuf, ni_o, no_o, ni_k, out, N);
}